// GATClassifier_12412455486159
// MI455X (gfx1250) — compile-verified
//
#include <hip/hip_runtime.h>
#include <math.h>

// ---------------------------------------------------------------------------
// GAT classifier for MI455X (gfx1250), f32 end-to-end.
// GEMMs use V_WMMA_F32_16X16X4_F32 (wave32 WMMA, exact f32 like the reference).
// Edge softmax/aggregation are HBM-bound scatter/gather with f32 atomics.
// ---------------------------------------------------------------------------

typedef __attribute__((ext_vector_type(2))) float v2f;
typedef __attribute__((ext_vector_type(8))) float v8f;

#define N_NODES 50000
#define TILE_M 64
#define TILE_N 64
#define TILE_K 32

// ---------------- utility: fill ----------------
__global__ void fill_f32(float* __restrict__ p, float v, long n) {
    long i = (long)blockIdx.x * blockDim.x + threadIdx.x;
    if (i < n) p[i] = v;
}

// ---------------- monotone float atomic max (init must be -inf) ----------------
__device__ __forceinline__ void atomicMaxF(float* addr, float val) {
    if (val >= 0.0f)
        atomicMax((int*)addr, __float_as_int(val));
    else
        atomicMin((unsigned int*)addr, (unsigned int)__float_as_int(val));
}

// ---------------- WMMA f32 GEMM: C[M,N] = A[M,K] @ B[K,N] ----------------
// A is fetched from up to 3 row-major segments (widths w0,w1,w2) so the
// concat([randfeat, func_emb, codebert]) never has to be materialized.
__global__ __launch_bounds__(256) void gemm_wmma_f32(
    const float* __restrict__ p0, const float* __restrict__ p1,
    const float* __restrict__ p2, int w0, int w1, int w2,
    const float* __restrict__ B, float* __restrict__ C,
    int M, int N, int K)
{
    __shared__ float As[TILE_M][TILE_K + 2];
    __shared__ float Bs[TILE_K][TILE_N + 2];

    const int tid  = threadIdx.x;
    const int lane = tid & 31;
    const int wave = tid >> 5;
    const int mt     = wave & 3;          // row-tile 0..3
    const int ntBase = (wave >> 2) * 2;   // col-tiles {0,1} or {2,3}

    const int blockRow = blockIdx.x * TILE_M;
    const int blockCol = blockIdx.y * TILE_N;

    v8f acc0 = {};
    v8f acc1 = {};

    const int mLocal = lane & 15;
    const int kOff   = (lane >> 4) << 1;  // lanes 0-15: K+0,1 ; lanes 16-31: K+2,3

    for (int kb = 0; kb < K; kb += TILE_K) {
        // --- stage A tile (64 x 32), 3-segment gather, zero-padded ---
#pragma unroll
        for (int i = 0; i < 8; ++i) {
            int idx = tid + i * 256;
            int m = idx >> 5;
            int k = idx & 31;
            int row = blockRow + m;
            int gk  = kb + k;
            float v = 0.0f;
            if (row < M && gk < K) {
                int kk = gk;
                if (kk < w0)               v = p0[(long)row * w0 + kk];
                else if ((kk -= w0) < w1)  v = p1[(long)row * w1 + kk];
                else                       v = p2[(long)row * w2 + (kk - w1)];
            }
            As[m][k] = v;
        }
        // --- stage B tile (32 x 64) ---
#pragma unroll
        for (int i = 0; i < 8; ++i) {
            int idx = tid + i * 256;
            int k = idx >> 6;
            int n = idx & 63;
            int gk  = kb + k;
            int col = blockCol + n;
            float v = 0.0f;
            if (gk < K && col < N) v = B[(long)gk * N + col];
            Bs[k][n] = v;
        }
        __syncthreads();

#pragma unroll
        for (int kk = 0; kk < TILE_K; kk += 4) {
            v2f a, b0, b1;
            // A fragment (16x4 f32): lane m = lane%16, VGPR0/1 = K kOff/kOff+1
            a.x = As[mt * 16 + mLocal][kk + kOff];
            a.y = As[mt * 16 + mLocal][kk + kOff + 1];
            // B fragment (4x16 f32): lane n = lane%16, VGPR0/1 = K kOff/kOff+1
            b0.x = Bs[kk + kOff][ntBase * 16 + mLocal];
            b0.y = Bs[kk + kOff + 1][ntBase * 16 + mLocal];
            b1.x = Bs[kk + kOff][(ntBase + 1) * 16 + mLocal];
            b1.y = Bs[kk + kOff + 1][(ntBase + 1) * 16 + mLocal];
            acc0 = __builtin_amdgcn_wmma_f32_16x16x4_f32(
                false, a, false, b0, (short)0, acc0, false, false);
            acc1 = __builtin_amdgcn_wmma_f32_16x16x4_f32(
                false, a, false, b1, (short)0, acc1, false, false);
        }
        __syncthreads();
    }

    // --- store C: VGPR r -> M = r (+8 for lanes 16-31), N = lane%16 ---
    const int col0 = blockCol + ntBase * 16 + mLocal;
    const int col1 = col0 + 16;
    const int mAdd = (lane >> 4) << 3;
#pragma unroll
    for (int r = 0; r < 8; ++r) {
        int row = blockRow + mt * 16 + r + mAdd;
        if (row < M) {
            C[(long)row * N + col0] = acc0[r];
            C[(long)row * N + col1] = acc1[r];
        }
    }
}

// ---------------- attention projections: el/er[n,h] = feat[n,h,:].al/ar[h,:] ----------------
__global__ __launch_bounds__(256) void attn_proj(
    const float* __restrict__ feat, const float* __restrict__ al,
    const float* __restrict__ ar, float* __restrict__ el,
    float* __restrict__ er, int nNodes, int H)
{
    int wid  = blockIdx.x * 8 + (threadIdx.x >> 5);
    int lane = threadIdx.x & 31;
    if (wid >= nNodes * H) return;
    int node = wid / H;
    int h    = wid - node * H;
    int C    = H * 64;
    const float* f  = feat + (long)node * C + h * 64;
    const float* wl = al + h * 64;
    const float* wr = ar + h * 64;
    float f0 = f[lane], f1 = f[lane + 32];
    float sl = f0 * wl[lane] + f1 * wl[lane + 32];
    float sr = f0 * wr[lane] + f1 * wr[lane + 32];
#pragma unroll
    for (int o = 16; o > 0; o >>= 1) {
        sl += __shfl_xor(sl, o, 32);
        sr += __shfl_xor(sr, o, 32);
    }
    if (lane == 0) { el[wid] = sl; er[wid] = sr; }
}

// ---------------- edge pass 1: e = leaky_relu(el[src]+er[dst]); segment max ----------------
__global__ void edge_lrelu_max(
    const int* __restrict__ src, const int* __restrict__ dst,
    const float* __restrict__ el, const float* __restrict__ er,
    float* __restrict__ ebuf, float* __restrict__ emax, int E, int H)
{
    int i = blockIdx.x * blockDim.x + threadIdx.x;
    if (i >= E) return;
    int s = src[i], d = dst[i];
    for (int h = 0; h < H; ++h) {
        float x = el[s * H + h] + er[d * H + h];
        float e = x > 0.0f ? x : 0.2f * x;
        ebuf[(long)i * H + h] = e;
        atomicMaxF(&emax[d * H + h], e);
    }
}

// ---------------- edge pass 2: ex = exp(e - emax[dst]); segment sum ----------------
__global__ void edge_expsum(
    const int* __restrict__ dst, float* __restrict__ ebuf,
    const float* __restrict__ emax, float* __restrict__ denom, int E, int H)
{
    int i = blockIdx.x * blockDim.x + threadIdx.x;
    if (i >= E) return;
    int d = dst[i];
    for (int h = 0; h < H; ++h) {
        float ex = expf(ebuf[(long)i * H + h] - emax[d * H + h]);
        ebuf[(long)i * H + h] = ex;
        atomicAdd(&denom[d * H + h], ex);
    }
}

// ---------------- edge pass 3: rst[dst] += feat[src] * alpha (wave per edge) ----------------
__global__ __launch_bounds__(256) void edge_aggregate(
    const int* __restrict__ src, const int* __restrict__ dst,
    const float* __restrict__ feat, const float* __restrict__ ebuf,
    const float* __restrict__ denom, float* __restrict__ rst, int E, int H)
{
    int wid  = blockIdx.x * 8 + (threadIdx.x >> 5);
    int lane = threadIdx.x & 31;
    if (wid >= E) return;
    int s = src[wid], d = dst[wid];
    int C = H * 64;
    for (int idx = lane; idx < C; idx += 32) {
        int h = idx >> 6;
        float alpha = ebuf[(long)wid * H + h] / denom[d * H + h];
        atomicAdd(&rst[(long)d * C + idx], feat[(long)s * C + idx] * alpha);
    }
}

// ---------------- h = elu(rst + bias), in place ----------------
__global__ void bias_elu(float* __restrict__ x, const float* __restrict__ b,
                         long n, int C)
{
    long i = (long)blockIdx.x * blockDim.x + threadIdx.x;
    if (i >= n) return;
    float v = x[i] + b[i % C];
    x[i] = v > 0.0f ? v : expm1f(v);
}

// ---------------- classifier: relu((rst2+b2)@Wc1+bc1) @ Wc2 + bc2 ----------------
__global__ __launch_bounds__(256) void classify(
    const float* __restrict__ rst2, const float* __restrict__ b2,
    const float* __restrict__ Wc1, const float* __restrict__ bc1,
    const float* __restrict__ Wc2, const float* __restrict__ bc2,
    float* __restrict__ out, int nNodes)
{
    int wid  = blockIdx.x * 8 + (threadIdx.x >> 5);
    int lane = threadIdx.x & 31;
    if (wid >= nNodes) return;
    float acc = 0.0f;
#pragma unroll
    for (int d = 0; d < 64; ++d)
        acc += (rst2[(long)wid * 64 + d] + b2[d]) * Wc1[d * 32 + lane];
    float t = fmaxf(acc + bc1[lane], 0.0f);
    float o0 = t * Wc2[lane * 2 + 0];
    float o1 = t * Wc2[lane * 2 + 1];
#pragma unroll
    for (int o = 16; o > 0; o >>= 1) {
        o0 += __shfl_xor(o0, o, 32);
        o1 += __shfl_xor(o1, o, 32);
    }
    if (lane == 0) {
        out[(long)wid * 2 + 0] = o0 + bc2[0];
        out[(long)wid * 2 + 1] = o1 + bc2[1];
    }
}

// ---------------------------------------------------------------------------
extern "C" void kernel_launch(void* const* d_in, const int* in_sizes, int n_in,
                              void* d_out, int out_size, void* d_ws, size_t ws_size,
                              hipStream_t stream)
{
    (void)n_in; (void)out_size; (void)ws_size;

    const float* randfeat = (const float*)d_in[0];
    const float* func_emb = (const float*)d_in[1];
    const float* codebert = (const float*)d_in[2];
    const int*   src      = (const int*)d_in[3];
    const int*   dst      = (const int*)d_in[4];
    const float* W1  = (const float*)d_in[5];
    const float* al1 = (const float*)d_in[6];
    const float* ar1 = (const float*)d_in[7];
    const float* b1  = (const float*)d_in[8];
    const float* W2  = (const float*)d_in[9];
    const float* al2 = (const float*)d_in[10];
    const float* ar2 = (const float*)d_in[11];
    const float* b2  = (const float*)d_in[12];
    const float* Wc1 = (const float*)d_in[13];
    const float* bc1 = (const float*)d_in[14];
    const float* Wc2 = (const float*)d_in[15];
    const float* bc2 = (const float*)d_in[16];
    float* out = (float*)d_out;

    const int  E  = in_sizes[3];
    const long NN = N_NODES;

    // workspace layout (floats)
    float* ws     = (float*)d_ws;
    float* feat1  = ws;                   // NN*256
    float* rst1   = feat1  + NN * 256;    // NN*256 (becomes h2 after bias+elu)
    float* feat2  = rst1   + NN * 256;    // NN*64
    float* rst2   = feat2  + NN * 64;     // NN*64
    float* el1    = rst2   + NN * 64;     // NN*4
    float* er1    = el1    + NN * 4;
    float* emax1  = er1    + NN * 4;
    float* denom1 = emax1  + NN * 4;
    float* el2    = denom1 + NN * 4;      // NN
    float* er2    = el2    + NN;
    float* emax2  = er2    + NN;
    float* denom2 = emax2  + NN;
    float* e1     = denom2 + NN;          // E*4
    float* e2     = e1 + (long)E * 4;     // E

    // ---- init accumulators every call (harness does not re-zero) ----
    fill_f32<<<(int)((NN * 256 + 255) / 256), 256, 0, stream>>>(rst1, 0.0f, NN * 256);
    fill_f32<<<(int)((NN * 64 + 255) / 256), 256, 0, stream>>>(rst2, 0.0f, NN * 64);
    fill_f32<<<(int)((NN * 4 + 255) / 256), 256, 0, stream>>>(denom1, 0.0f, NN * 4);
    fill_f32<<<(int)((NN + 255) / 256), 256, 0, stream>>>(denom2, 0.0f, NN);
    fill_f32<<<(int)((NN * 4 + 255) / 256), 256, 0, stream>>>(emax1, -INFINITY, NN * 4);
    fill_f32<<<(int)((NN + 255) / 256), 256, 0, stream>>>(emax2, -INFINITY, NN);

    // ---- layer 1 ----
    dim3 g1((N_NODES + TILE_M - 1) / TILE_M, 256 / TILE_N);
    gemm_wmma_f32<<<g1, 256, 0, stream>>>(randfeat, func_emb, codebert,
                                          100, 100, 768, W1, feat1,
                                          N_NODES, 256, 968);
    attn_proj<<<(N_NODES * 4 + 7) / 8, 256, 0, stream>>>(feat1, al1, ar1, el1, er1, N_NODES, 4);
    edge_lrelu_max<<<(E + 255) / 256, 256, 0, stream>>>(src, dst, el1, er1, e1, emax1, E, 4);
    edge_expsum<<<(E + 255) / 256, 256, 0, stream>>>(dst, e1, emax1, denom1, E, 4);
    edge_aggregate<<<(E + 7) / 8, 256, 0, stream>>>(src, dst, feat1, e1, denom1, rst1, E, 4);
    bias_elu<<<(int)((NN * 256 + 255) / 256), 256, 0, stream>>>(rst1, b1, NN * 256, 256);

    // ---- layer 2 ----
    dim3 g2((N_NODES + TILE_M - 1) / TILE_M, 64 / TILE_N);
    gemm_wmma_f32<<<g2, 256, 0, stream>>>(rst1, rst1, rst1,
                                          256, 0, 0, W2, feat2,
                                          N_NODES, 64, 256);
    attn_proj<<<(N_NODES + 7) / 8, 256, 0, stream>>>(feat2, al2, ar2, el2, er2, N_NODES, 1);
    edge_lrelu_max<<<(E + 255) / 256, 256, 0, stream>>>(src, dst, el2, er2, e2, emax2, E, 1);
    edge_expsum<<<(E + 255) / 256, 256, 0, stream>>>(dst, e2, emax2, denom2, E, 1);
    edge_aggregate<<<(E + 7) / 8, 256, 0, stream>>>(src, dst, feat2, e2, denom2, rst2, E, 1);

    // ---- classifier head ----
    classify<<<(N_NODES + 7) / 8, 256, 0, stream>>>(rst2, b2, Wc1, bc1, Wc2, bc2, out, N_NODES);
}